// FourPosFusionEmbedding_31379031064638
// MI455X (gfx1250) — compile-verified
//
#include <hip/hip_runtime.h>

#define MAXSEQ 512
#define TBL    1025          // 2*MAXSEQ + 1
#define NH     12
#define BATCH  4
#define SEQ    512

typedef __attribute__((ext_vector_type(2))) float v2f;
typedef __attribute__((ext_vector_type(8))) float v8f;
typedef int v4i __attribute__((vector_size(16)));      // matches builtin param type

typedef __attribute__((address_space(1))) v4i* gptr_v4i;   // global
typedef __attribute__((address_space(3))) v4i* lptr_v4i;   // LDS

#if __has_builtin(__builtin_amdgcn_global_load_async_to_lds_b128) && \
    __has_builtin(__builtin_amdgcn_s_wait_asynccnt)
#define HAVE_ASYNC_LDS 1
#else
#define HAVE_ASYNC_LDS 0
#endif

// ---------------------------------------------------------------------------
// Stage 1: P[t][row][h] = sum_k pe_t[row][k] * W[t*12 + k][h]
// One wave per 16-row tile per table: 4 tables * 65 tiles = 260 waves.
// V_WMMA_F32_16X16X4_F32, K=12 as 3 accumulating steps. All loads are
// unconditional (clamped addresses) so the wave stays convergent (EXEC all-1s)
// through the WMMA sequence; only the final stores are predicated.
// ---------------------------------------------------------------------------
__global__ void __launch_bounds__(32) pos_proj_wmma(
    const float* __restrict__ pe_ss, const float* __restrict__ pe_se,
    const float* __restrict__ pe_es, const float* __restrict__ pe_ee,
    const float* __restrict__ W, float* __restrict__ P)
{
    const int t  = blockIdx.x / 65;
    const int rt = blockIdx.x % 65;
    const float* pe = (t == 0) ? pe_ss : (t == 1) ? pe_se : (t == 2) ? pe_es : pe_ee;

    const int lane = threadIdx.x;        // 0..31
    const int half = lane >> 4;          // K sub-pair select
    const int m    = lane & 15;          // A row (M) / B column (N)
    const int row  = rt * 16 + m;
    const int rowc = row < TBL ? row : (TBL - 1);   // branchless tail clamp
    const int mc   = m < NH ? m : (NH - 1);         // branchless column clamp
    const float nmask = (m < NH) ? 1.0f : 0.0f;     // zero padded B columns

    v8f acc = {};
#pragma unroll
    for (int s = 0; s < 3; ++s) {
        const int k0 = s * 4 + half * 2;            // K base for this lane half
        v2f a;
        a.x = pe[rowc * NH + k0];
        a.y = pe[rowc * NH + k0 + 1];
        v2f bm;
        bm.x = W[(t * NH + k0) * NH + mc] * nmask;
        bm.y = W[(t * NH + k0 + 1) * NH + mc] * nmask;
        acc = __builtin_amdgcn_wmma_f32_16x16x4_f32(false, a, false, bm,
                                                    (short)0, acc, false, false);
    }

#pragma unroll
    for (int r = 0; r < 8; ++r) {
        const int orow = rt * 16 + half * 8 + r;    // D row for this VGPR/lane-half
        if (m < NH && orow < TBL)
            P[(t * TBL + orow) * NH + m] = acc[r];
    }
}

// ---------------------------------------------------------------------------
// Stage 2: out[b][h][i][j] = relu( sum_t P[t][idx_t(i,j)][h] + bias[h] )
// Projected tables (192.2 KB) live in dynamic LDS (320 KB/WGP cap), broadcast
// via GLOBAL_LOAD_ASYNC_TO_LDS_B128 (ASYNCcnt-tracked, bypasses VGPRs) when
// the toolchain exposes it. Grid: 4 batches x 32 i-tiles (16 rows each).
// 256 threads = 8 waves; lane == j for coalesced 128B store lines; gathers are
// 3x ds_load_b128 per table (48 B rows are 16 B aligned).
// ---------------------------------------------------------------------------
__global__ void __launch_bounds__(256) pos_fuse(
    const float* __restrict__ P, const int* __restrict__ pos_s,
    const int* __restrict__ pos_e, const float* __restrict__ bias,
    float* __restrict__ out)
{
    extern __shared__ float smem[];
    float* Tl = smem;                               // 4*1025*12 = 49200 floats
    int*   ps = (int*)(smem + 4 * TBL * NH);        // 512 ints
    int*   pE = ps + SEQ;                           // 512 ints

    const int b   = blockIdx.x >> 5;                // batch
    const int it  = blockIdx.x & 31;                // i-tile (16 rows)
    const int tid = threadIdx.x;

    // ---- broadcast projected tables into LDS ----
    const float4* Pv = (const float4*)P;
    float4* Tv = (float4*)Tl;
#if HAVE_ASYNC_LDS
    for (int k = tid; k < (4 * TBL * NH) / 4; k += 256) {
        __builtin_amdgcn_global_load_async_to_lds_b128(
            (gptr_v4i)(Pv + k),
            (lptr_v4i)(Tv + k),
            /*offset=*/0, /*cpol=*/0);
    }
#else
    for (int k = tid; k < (4 * TBL * NH) / 4; k += 256) Tv[k] = Pv[k];
#endif
    // pos arrays + bias overlap the bulk async copy
    for (int k = tid; k < SEQ; k += 256) {
        ps[k] = pos_s[b * SEQ + k];
        pE[k] = pos_e[b * SEQ + k];
    }
    float breg[NH];
#pragma unroll
    for (int h = 0; h < NH; ++h) breg[h] = bias[h];   // broadcast, L2/K$ resident
#if HAVE_ASYNC_LDS
    __builtin_amdgcn_s_wait_asynccnt(0);
#endif
    __syncthreads();

    const int lane = tid & 31;
    const int w    = tid >> 5;                      // wave id 0..7

#pragma unroll
    for (int ii = 0; ii < 2; ++ii) {
        const int i   = it * 16 + ii * 8 + w;
        const int psi = ps[i];
        const int pei = pE[i];
        for (int j0 = 0; j0 < SEQ; j0 += 32) {
            const int j   = j0 + lane;
            const int psj = ps[j];
            const int pej = pE[j];
            const float4* tss = (const float4*)(Tl + 0 * TBL * NH + (psi - psj + MAXSEQ) * NH);
            const float4* tse = (const float4*)(Tl + 1 * TBL * NH + (psi - pej + MAXSEQ) * NH);
            const float4* tes = (const float4*)(Tl + 2 * TBL * NH + (pei - psj + MAXSEQ) * NH);
            const float4* tee = (const float4*)(Tl + 3 * TBL * NH + (pei - pej + MAXSEQ) * NH);

            float acc[NH];
#pragma unroll
            for (int q = 0; q < 3; ++q) {
                float4 x0 = tss[q], x1 = tse[q], x2 = tes[q], x3 = tee[q];
                acc[4 * q + 0] = x0.x + x1.x + x2.x + x3.x;
                acc[4 * q + 1] = x0.y + x1.y + x2.y + x3.y;
                acc[4 * q + 2] = x0.z + x1.z + x2.z + x3.z;
                acc[4 * q + 3] = x0.w + x1.w + x2.w + x3.w;
            }

            const size_t base = (size_t)b * NH * SEQ * SEQ + (size_t)i * SEQ + j;
#pragma unroll
            for (int h = 0; h < NH; ++h) {
                float v = acc[h] + breg[h];
                out[base + (size_t)h * SEQ * SEQ] = v > 0.0f ? v : 0.0f;
            }
        }
    }
}

// ---------------------------------------------------------------------------
extern "C" void kernel_launch(void* const* d_in, const int* in_sizes, int n_in,
                              void* d_out, int out_size, void* d_ws, size_t ws_size,
                              hipStream_t stream)
{
    const int*   pos_s = (const int*)d_in[0];
    const int*   pos_e = (const int*)d_in[1];
    const float* pe_ss = (const float*)d_in[2];
    const float* pe_se = (const float*)d_in[3];
    const float* pe_es = (const float*)d_in[4];
    const float* pe_ee = (const float*)d_in[5];
    const float* W     = (const float*)d_in[6];
    const float* bias  = (const float*)d_in[7];
    float* out = (float*)d_out;
    float* P   = (float*)d_ws;          // 4*1025*12 floats = 196,800 bytes scratch

    // Stage 1: tiny WMMA GEMM projecting each table through its W slice.
    pos_proj_wmma<<<4 * 65, 32, 0, stream>>>(pe_ss, pe_se, pe_es, pe_ee, W, P);

    // Stage 2: gather-sum-relu from LDS-resident projected tables.
    const size_t smem_bytes = (size_t)(4 * TBL * NH) * sizeof(float)  // 196,800
                            + (size_t)(2 * SEQ) * sizeof(int);         //   4,096
    pos_fuse<<<BATCH * 32, 256, smem_bytes, stream>>>(P, pos_s, pos_e, bias, out);
}